// GIN_Net_15015205667099
// MI455X (gfx1250) — compile-verified
//
#include <hip/hip_runtime.h>

#define N_NODES 100000
#define N_EDGES 1250000
#define D 64
#define BN_EPS 1e-5f
#define LDA 65  // LDS row stride (floats) to spread banks

typedef float v2f __attribute__((ext_vector_type(2)));
typedef float v8f __attribute__((ext_vector_type(8)));

// ---------------------------------------------------------------- zero
__global__ void k_zero(float* __restrict__ p, long n) {
    long i = (long)blockIdx.x * blockDim.x + threadIdx.x;
    if (i < n) p[i] = 0.0f;
}

// ---------------------------------------------------------------- scatter-add
// agg[dst[e]] += x[src[e]]  ; 16 threads per edge, float4 per thread
__global__ void k_scatter(const float* __restrict__ x,
                          const int* __restrict__ src,
                          const int* __restrict__ dst,
                          float* __restrict__ agg, int E) {
    int tid = blockIdx.x * blockDim.x + threadIdx.x;
    int e = tid >> 4;
    if (e >= E) return;
    int c = (tid & 15) << 2;
    int s = src[e];
    int d = dst[e];
    const float4 v = *(const float4*)(x + (long)s * D + c);
    float* p = agg + (long)d * D + c;
    atomicAdd(p + 0, v.x);
    atomicAdd(p + 1, v.y);
    atomicAdd(p + 2, v.z);
    atomicAdd(p + 3, v.w);
}

// ---------------------------------------------------------------- fused GEMM
// out = f(A0 [+ A1]) @ W + bias ; f = identity or BN+ReLU (applied on load)
// Also accumulates per-column sum / sumsq of `out` into gsum/gsumsq.
// Block: 128 threads = 4 waves; 64 rows per block; wave handles 16x64 tile.
__global__ __launch_bounds__(128) void k_gemm(
    const float* __restrict__ A0, const float* __restrict__ A1,
    const float* __restrict__ W, const float* __restrict__ bias,
    const float* __restrict__ mean, const float* __restrict__ rstd,
    const float* __restrict__ gamma, const float* __restrict__ beta,
    int applyBn,
    float* __restrict__ out, float* __restrict__ gsum,
    float* __restrict__ gsumsq, int M) {
    __shared__ float lsA[64 * LDA];
    __shared__ float lsW[64 * LDA];
    __shared__ float lsSum[D];
    __shared__ float lsSq[D];

    const int t = threadIdx.x;
    const int blockRow = blockIdx.x * 64;

    // stage weights (64x64, row-major: W[k][n])
    for (int i = t; i < 64 * 64; i += 128) {
        int r = i >> 6, c = i & 63;
        lsW[r * LDA + c] = W[i];
    }
    // stage A tile with optional addend + BN/ReLU transform
    for (int i = t; i < 64 * 64; i += 128) {
        int r = i >> 6, c = i & 63;
        int gr = blockRow + r;
        if (gr >= M) gr = M - 1;  // clamp (stores are guarded later)
        long off = (long)gr * D + c;
        float v = A0[off];
        if (A1) v += A1[off];
        if (applyBn) {
            v = (v - mean[c]) * rstd[c] * gamma[c] + beta[c];
            v = fmaxf(v, 0.0f);
        }
        lsA[r * LDA + c] = v;
    }
    if (t < D) { lsSum[t] = 0.0f; lsSq[t] = 0.0f; }
    __syncthreads();

    const int wave = t >> 5;
    const int lane = t & 31;
    const int m    = lane & 15;   // row-in-tile (A) / col-in-16 (B,C)
    const int hi   = lane >> 4;   // lane half
    const int arow = wave * 16 + m;

    v8f acc[4];
#pragma unroll
    for (int nt = 0; nt < 4; ++nt) {
        float bv = bias[nt * 16 + m];
        acc[nt] = (v8f){bv, bv, bv, bv, bv, bv, bv, bv};
    }

#pragma unroll
    for (int k = 0; k < 16; ++k) {
        // A 16x4 f32: lanes 0-15 K={0,1}, lanes 16-31 K={2,3}
        v2f a;
        a.x = lsA[arow * LDA + k * 4 + hi * 2 + 0];
        a.y = lsA[arow * LDA + k * 4 + hi * 2 + 1];
        const int krow = k * 4 + hi * 2;
#pragma unroll
        for (int nt = 0; nt < 4; ++nt) {
            // B 4x16 f32: vgpr0 = rows {0,2}, vgpr1 = rows {1,3}
            v2f b;
            b.x = lsW[(krow + 0) * LDA + nt * 16 + m];
            b.y = lsW[(krow + 1) * LDA + nt * 16 + m];
            acc[nt] = __builtin_amdgcn_wmma_f32_16x16x4_f32(
                false, a, false, b, (short)0, acc[nt], false, false);
        }
    }

    // store D-matrix (C vgpr i: rows i and i+8) + column stats
#pragma unroll
    for (int nt = 0; nt < 4; ++nt) {
        const int col = nt * 16 + m;
        float s = 0.0f, s2 = 0.0f;
#pragma unroll
        for (int i = 0; i < 8; ++i) {
            int gr = blockRow + wave * 16 + hi * 8 + i;
            float v = acc[nt][i];
            if (gr < M) {
                out[(long)gr * D + col] = v;
                s += v;
                s2 += v * v;
            }
        }
        atomicAdd(&lsSum[col], s);
        atomicAdd(&lsSq[col], s2);
    }
    __syncthreads();
    if (t < D) {
        atomicAdd(&gsum[t], lsSum[t]);
        atomicAdd(&gsumsq[t], lsSq[t]);
    }
}

// ---------------------------------------------------------------- BN finalize
__global__ void k_bnstat(const float* __restrict__ gsum,
                         const float* __restrict__ gsumsq,
                         float* __restrict__ mean, float* __restrict__ rstd,
                         int M) {
    int t = threadIdx.x;  // 64 threads
    float mu  = gsum[t] / (float)M;
    float var = gsumsq[t] / (float)M - mu * mu;
    mean[t] = mu;
    rstd[t] = rsqrtf(var + BN_EPS);
}

// ---------------------------------------------------------------- BN apply
__global__ void k_apply(const float* __restrict__ in,
                        const float* __restrict__ mean,
                        const float* __restrict__ rstd,
                        const float* __restrict__ g,
                        const float* __restrict__ be,
                        float* __restrict__ out, int relu, long n) {
    long i = (long)blockIdx.x * blockDim.x + threadIdx.x;
    if (i >= n) return;
    int c = (int)(i & 63);
    float v = (in[i] - mean[c]) * rstd[c] * g[c] + be[c];
    if (relu) v = fmaxf(v, 0.0f);
    out[i] = v;
}

// ---------------------------------------------------------------- launch
extern "C" void kernel_launch(void* const* d_in, const int* in_sizes, int n_in,
                              void* d_out, int out_size, void* d_ws,
                              size_t ws_size, hipStream_t stream) {
    const float* x   = (const float*)d_in[0];
    const int*   ei  = (const int*)d_in[1];
    const int*   src = ei;
    const int*   dst = ei + N_EDGES;
    const float* w1a = (const float*)d_in[2];
    const float* w1b = (const float*)d_in[3];
    const float* w2a = (const float*)d_in[4];
    const float* w2b = (const float*)d_in[5];
    const float* b1a = (const float*)d_in[6];
    const float* b1b = (const float*)d_in[7];
    const float* b2a = (const float*)d_in[8];
    const float* b2b = (const float*)d_in[9];
    const float* be1a = (const float*)d_in[10];
    const float* be1b = (const float*)d_in[11];
    const float* be2a = (const float*)d_in[12];
    const float* be2b = (const float*)d_in[13];
    const float* g1a = (const float*)d_in[14];
    const float* g1b = (const float*)d_in[15];
    const float* g2a = (const float*)d_in[16];
    const float* g2b = (const float*)d_in[17];

    const long ND = (long)N_NODES * D;
    float* ws    = (float*)d_ws;
    float* bufA  = ws;            // agg / t4
    float* bufB  = ws + ND;       // t1 / h1
    float* bufC  = ws + 2 * ND;   // t2 / t3
    float* sum   = ws + 3 * ND;
    float* sumsq = sum + 64;
    float* mean  = sum + 128;
    float* rstd  = sum + 192;

    const int gemmGrid    = (N_NODES + 63) / 64;
    const int zeroGridND  = (int)((ND + 255) / 256);
    const int scatterGrid = (int)(((long)N_EDGES * 16 + 255) / 256);
    float* outp = (float*)d_out;

    // ---------------- layer 1 ----------------
    k_zero<<<zeroGridND, 256, 0, stream>>>(bufA, ND);
    k_zero<<<1, 128, 0, stream>>>(sum, 128);
    k_scatter<<<scatterGrid, 256, 0, stream>>>(x, src, dst, bufA, N_EDGES);
    // t1 = (x + agg) @ w1a + b1a   (+ stats of t1)
    k_gemm<<<gemmGrid, 128, 0, stream>>>(x, bufA, w1a, b1a, mean, rstd, g1a,
                                         be1a, 0, bufB, sum, sumsq, N_NODES);
    k_bnstat<<<1, 64, 0, stream>>>(sum, sumsq, mean, rstd, N_NODES);
    // t2 = relu(bn(t1)) @ w1b + b1b   (+ stats of t2)
    k_zero<<<1, 128, 0, stream>>>(sum, 128);
    k_gemm<<<gemmGrid, 128, 0, stream>>>(bufB, nullptr, w1b, b1b, mean, rstd,
                                         g1a, be1a, 1, bufC, sum, sumsq,
                                         N_NODES);
    k_bnstat<<<1, 64, 0, stream>>>(sum, sumsq, mean, rstd, N_NODES);
    // h1 = relu(bn(t2))  (BN of t2 then inter-layer ReLU)
    k_apply<<<zeroGridND, 256, 0, stream>>>(bufC, mean, rstd, g1b, be1b, bufB,
                                            1, ND);

    // ---------------- layer 2 ----------------
    k_zero<<<zeroGridND, 256, 0, stream>>>(bufA, ND);
    k_zero<<<1, 128, 0, stream>>>(sum, 128);
    k_scatter<<<scatterGrid, 256, 0, stream>>>(bufB, src, dst, bufA, N_EDGES);
    // t3 = (h1 + agg) @ w2a + b2a
    k_gemm<<<gemmGrid, 128, 0, stream>>>(bufB, bufA, w2a, b2a, mean, rstd, g2a,
                                         be2a, 0, bufC, sum, sumsq, N_NODES);
    k_bnstat<<<1, 64, 0, stream>>>(sum, sumsq, mean, rstd, N_NODES);
    // t4 = relu(bn(t3)) @ w2b + b2b
    k_zero<<<1, 128, 0, stream>>>(sum, 128);
    k_gemm<<<gemmGrid, 128, 0, stream>>>(bufC, nullptr, w2b, b2b, mean, rstd,
                                         g2a, be2a, 1, bufA, sum, sumsq,
                                         N_NODES);
    k_bnstat<<<1, 64, 0, stream>>>(sum, sumsq, mean, rstd, N_NODES);
    // out = bn(t4)   (no relu)
    k_apply<<<zeroGridND, 256, 0, stream>>>(bufA, mean, rstd, g2b, be2b, outp,
                                            0, ND);
}